// GNN_82884278878945
// MI455X (gfx1250) — compile-verified
//
#include <hip/hip_runtime.h>
#include <cmath>

// ---------------------------------------------------------------------------
// Types for CDNA5 WMMA f32 16x16x4:
//   A 16x4  f32 -> 64 vals / 32 lanes = 2 VGPRs (float2)
//   B 4x16  f32 -> 2 VGPRs
//   C/D 16x16 f32 -> 8 VGPRs
// ---------------------------------------------------------------------------
typedef __attribute__((ext_vector_type(2))) float v2f;
typedef __attribute__((ext_vector_type(8))) float v8f;

// ---------------------------------------------------------------------------
// Helpers
// ---------------------------------------------------------------------------
__device__ __forceinline__ void atomicMaxFloat(float* addr, float val) {
    // monotone int mapping trick; buffer must be initialized to -inf
    unsigned int bits = __float_as_uint(val);
    if ((bits >> 31) == 0u) {
        atomicMax((int*)addr, (int)bits);           // val >= +0
    } else {
        atomicMin((unsigned int*)addr, bits);       // val < 0 (or -0)
    }
}

__global__ void fill_kernel(float* __restrict__ p, float v, size_t n) {
    size_t i = (size_t)blockIdx.x * blockDim.x + threadIdx.x;
    if (i < n) p[i] = v;
}

// ---------------------------------------------------------------------------
// GEMM: Y[rows,128] = (A * scale + shift) @ W[128,128] + bias
// One block = 8 waves; wave w computes the 16x16 tile at cols [16w,16w+16).
// A tile (16x128 f32, 8KB) staged in LDS; K-loop of 32 x WMMA f32 16x16x4.
// rows must be a multiple of 16 (N=50000 = 3125*16).
// ---------------------------------------------------------------------------
__global__ __launch_bounds__(256)
void gemm128_wmma(const float* __restrict__ A, const float* __restrict__ W,
                  const float* __restrict__ bias,
                  const float* __restrict__ scale, const float* __restrict__ shift,
                  float* __restrict__ Y) {
    __shared__ float sA[16 * 128];

    const int rowBase = blockIdx.x * 16;
    const float* Ab = A + (size_t)rowBase * 128;

    // cooperative load of the 16x128 A tile, with optional BN scale/shift fused
    for (int i = threadIdx.x; i < 16 * 128; i += 256) {
        float v = Ab[i];
        if (scale) {
            int ch = i & 127;
            v = v * scale[ch] + shift[ch];
        }
        sA[i] = v;
    }
    __syncthreads();

    const int lane    = threadIdx.x & 31;
    const int wv      = threadIdx.x >> 5;      // 0..7 -> column tile
    const int colBase = wv * 16;
    const int mr      = lane & 15;             // A row for this lane
    const int koff    = (lane >> 4) << 1;      // 0 (lanes 0-15) or 2 (lanes 16-31)
    const int nc      = lane & 15;             // B column for this lane

    const float* sArow = sA + mr * 128 + koff;
    const float* Wp    = W + (size_t)koff * 128 + colBase + nc;

    v8f acc = {};
    #pragma unroll
    for (int k = 0; k < 128; k += 4) {
        v2f a, b;
        a.x = sArow[k];
        a.y = sArow[k + 1];
        b.x = Wp[(size_t)k * 128];
        b.y = Wp[(size_t)(k + 1) * 128];
        acc = __builtin_amdgcn_wmma_f32_16x16x4_f32(
            /*neg_a=*/false, a, /*neg_b=*/false, b,
            /*c_mod=*/(short)0, acc, /*reuse_a=*/false, /*reuse_b=*/false);
    }

    // C/D layout: VGPR r -> row (lane<16 ? r : 8+r), col = lane&15
    const int crow = (lane >> 4) << 3;         // 0 or 8
    const float bv = bias ? bias[colBase + nc] : 0.0f;
    float* Yp = Y + (size_t)(rowBase + crow) * 128 + colBase + nc;
    #pragma unroll
    for (int r = 0; r < 8; ++r) {
        Yp[(size_t)r * 128] = acc[r] + bv;
    }
}

// ---------------------------------------------------------------------------
// Edge phase 1: attention logits + segment max.
// One thread per (edge, head); self-loop edges are t in [E, E+N).
// ---------------------------------------------------------------------------
__global__ __launch_bounds__(256)
void edge_logits_kernel(const int* __restrict__ ei, const float* __restrict__ xl,
                        const float* __restrict__ xr, const float* __restrict__ att,
                        float* __restrict__ logits, float* __restrict__ maxbuf,
                        int E, int Et) {
    int gid = blockIdx.x * blockDim.x + threadIdx.x;
    if (gid >= Et * 8) return;
    int t = gid >> 3, h = gid & 7;
    int s, d;
    if (t < E) { s = ei[t]; d = ei[E + t]; } else { s = t - E; d = s; }

    const float4* xls = (const float4*)(xl + (size_t)s * 128 + h * 16);
    const float4* xrd = (const float4*)(xr + (size_t)d * 128 + h * 16);
    const float4* at4 = (const float4*)(att + h * 16);

    float lg = 0.f;
    #pragma unroll
    for (int j = 0; j < 4; ++j) {
        float4 a = xls[j], b = xrd[j], w = at4[j];
        float vx = a.x + b.x, vy = a.y + b.y, vz = a.z + b.z, vw = a.w + b.w;
        vx = vx > 0.f ? vx : 0.2f * vx;
        vy = vy > 0.f ? vy : 0.2f * vy;
        vz = vz > 0.f ? vz : 0.2f * vz;
        vw = vw > 0.f ? vw : 0.2f * vw;
        lg += w.x * vx + w.y * vy + w.z * vz + w.w * vw;
    }
    logits[gid] = lg;
    atomicMaxFloat(maxbuf + (size_t)d * 8 + h, lg);
}

// ---------------------------------------------------------------------------
// Edge phase 2: exp(logit - max) + segment sum (denominator).
// ---------------------------------------------------------------------------
__global__ __launch_bounds__(256)
void edge_exp_kernel(const int* __restrict__ ei, float* __restrict__ logits,
                     const float* __restrict__ maxbuf, float* __restrict__ denom,
                     int E, int Et) {
    int gid = blockIdx.x * blockDim.x + threadIdx.x;
    if (gid >= Et * 8) return;
    int t = gid >> 3, h = gid & 7;
    int d = (t < E) ? ei[E + t] : (t - E);
    float ex = expf(logits[gid] - maxbuf[(size_t)d * 8 + h]);
    logits[gid] = ex;
    unsafeAtomicAdd(denom + (size_t)d * 8 + h, ex);
}

// ---------------------------------------------------------------------------
// Edge phase 3: alpha = ex/denom; scatter-add alpha * xl[src] into Hacc[dst].
// ---------------------------------------------------------------------------
__global__ __launch_bounds__(256)
void edge_scatter_kernel(const int* __restrict__ ei, const float* __restrict__ xl,
                         const float* __restrict__ exbuf, const float* __restrict__ denom,
                         float* __restrict__ Hacc, int E, int Et) {
    int gid = blockIdx.x * blockDim.x + threadIdx.x;
    if (gid >= Et * 8) return;
    int t = gid >> 3, h = gid & 7;
    int s, d;
    if (t < E) { s = ei[t]; d = ei[E + t]; } else { s = t - E; d = s; }

    float alpha = exbuf[gid] / (denom[(size_t)d * 8 + h] + 1e-16f);
    const float4* xs = (const float4*)(xl + (size_t)s * 128 + h * 16);
    float* out = Hacc + (size_t)d * 128 + h * 16;
    #pragma unroll
    for (int j = 0; j < 4; ++j) {
        float4 v = xs[j];
        unsafeAtomicAdd(out + 4 * j + 0, alpha * v.x);
        unsafeAtomicAdd(out + 4 * j + 1, alpha * v.y);
        unsafeAtomicAdd(out + 4 * j + 2, alpha * v.z);
        unsafeAtomicAdd(out + 4 * j + 3, alpha * v.w);
    }
}

// ---------------------------------------------------------------------------
// relu(H + conv_bias) in place + per-channel sum / sumsq for BatchNorm.
// 128 threads = 1 channel each; blocks stride over rows (coalesced).
// ---------------------------------------------------------------------------
__global__ __launch_bounds__(128)
void relu_bn_stats_kernel(float* __restrict__ H, const float* __restrict__ bias,
                          float* __restrict__ sum, float* __restrict__ sumsq, int N) {
    int ch = threadIdx.x;
    float b = bias[ch];
    float s = 0.f, ss = 0.f;
    for (int r = blockIdx.x; r < N; r += gridDim.x) {
        size_t idx = (size_t)r * 128 + ch;
        float v = H[idx] + b;
        v = fmaxf(v, 0.f);
        H[idx] = v;
        s += v;
        ss += v * v;
    }
    unsafeAtomicAdd(sum + ch, s);
    unsafeAtomicAdd(sumsq + ch, ss);
}

__global__ void bn_final_kernel(const float* __restrict__ sum, const float* __restrict__ sumsq,
                                const float* __restrict__ gamma, const float* __restrict__ beta,
                                float* __restrict__ scale, float* __restrict__ shift, int N) {
    int c = threadIdx.x;   // 128 threads
    float invN = 1.0f / (float)N;
    float mean = sum[c] * invN;
    float var  = sumsq[c] * invN - mean * mean;
    float sc   = gamma[c] / sqrtf(var + 1e-5f);
    scale[c] = sc;
    shift[c] = beta[c] - mean * sc;
}

// ---------------------------------------------------------------------------
// Edge decoder: 8 lanes per edge, each lane handles 16 dims, shfl-xor reduce.
// value1 = sigmoid(-(relu(a)*dist + b)) = 1/(1+exp(relu(a)*dist + b))
// ---------------------------------------------------------------------------
__global__ __launch_bounds__(256)
void edge_dec_kernel(const int* __restrict__ ei, const float* __restrict__ z,
                     const float* __restrict__ a_e, const float* __restrict__ b_e,
                     float* __restrict__ val1, int E) {
    int gid = blockIdx.x * blockDim.x + threadIdx.x;
    if (gid >= E * 8) return;
    int e = gid >> 3, j = gid & 7;
    int s = ei[e], d = ei[E + e];

    const float4* zs = (const float4*)(z + (size_t)s * 128 + j * 16);
    const float4* zd = (const float4*)(z + (size_t)d * 128 + j * 16);
    float p = 0.f;
    #pragma unroll
    for (int q = 0; q < 4; ++q) {
        float4 a = zs[q], b = zd[q];
        float dx = a.x - b.x, dy = a.y - b.y, dz = a.z - b.z, dw = a.w - b.w;
        p += dx * dx + dy * dy + dz * dz + dw * dw;
    }
    p += __shfl_xor(p, 1, 8);
    p += __shfl_xor(p, 2, 8);
    p += __shfl_xor(p, 4, 8);
    if (j == 0) {
        float dist = fmaxf(a_e[0], 0.f) * p + b_e[0];
        val1[e] = 1.0f / (1.0f + expf(dist));
    }
}

// ---------------------------------------------------------------------------
// Launch
// ---------------------------------------------------------------------------
extern "C" void kernel_launch(void* const* d_in, const int* in_sizes, int n_in,
                              void* d_out, int out_size, void* d_ws, size_t ws_size,
                              hipStream_t stream) {
    const float* x         = (const float*)d_in[0];
    const int*   ei        = (const int*)d_in[1];
    const float* W_l       = (const float*)d_in[2];
    const float* b_l       = (const float*)d_in[3];
    const float* W_r       = (const float*)d_in[4];
    const float* b_r       = (const float*)d_in[5];
    const float* att       = (const float*)d_in[6];
    const float* conv_bias = (const float*)d_in[7];
    const float* bn_gamma  = (const float*)d_in[8];
    const float* bn_beta   = (const float*)d_in[9];
    const float* lin_W     = (const float*)d_in[10];
    const float* lin_b     = (const float*)d_in[11];
    const float* a_edge    = (const float*)d_in[12];
    const float* b_edge    = (const float*)d_in[13];
    const float* dec2_W    = (const float*)d_in[14];
    const float* dec2_b    = (const float*)d_in[15];

    const int N  = in_sizes[0] / 128;   // 50000
    const int E  = in_sizes[1] / 2;     // 800000
    const int Et = E + N;               // with self-loops

    // workspace layout (floats)
    float* ws     = (float*)d_ws;
    float* xl     = ws;                             // N*128
    float* xr     = xl + (size_t)N * 128;           // N*128
    float* Hacc   = xr + (size_t)N * 128;           // N*128 (becomes Hfeat)
    float* exbuf  = Hacc + (size_t)N * 128;         // Et*8  (logits -> exp)
    float* maxbuf = exbuf + (size_t)Et * 8;         // N*8
    float* denom  = maxbuf + (size_t)N * 8;         // N*8
    float* stats  = denom + (size_t)N * 8;          // 512
    float* sumb = stats, *sumsq = stats + 128, *scaleb = stats + 256, *shiftb = stats + 384;

    // output layout: z [N*128], value1 [E], value2 [N*128]
    float* z    = (float*)d_out;
    float* val1 = z + (size_t)N * 128;
    float* val2 = val1 + E;

    // ---- init accumulators (deterministic every call) ----
    size_t nH = (size_t)N * 128;
    fill_kernel<<<(unsigned)((nH + 255) / 256), 256, 0, stream>>>(Hacc, 0.f, nH);
    fill_kernel<<<(N * 8 + 255) / 256, 256, 0, stream>>>(maxbuf, -INFINITY, (size_t)N * 8);
    fill_kernel<<<(N * 8 + 255) / 256, 256, 0, stream>>>(denom, 0.f, (size_t)N * 8);
    fill_kernel<<<2, 256, 0, stream>>>(stats, 0.f, 512);

    const int rowTiles = N / 16;  // 3125

    // ---- xl = x@W_l + b_l ; xr = x@W_r + b_r ----
    gemm128_wmma<<<rowTiles, 256, 0, stream>>>(x, W_l, b_l, nullptr, nullptr, xl);
    gemm128_wmma<<<rowTiles, 256, 0, stream>>>(x, W_r, b_r, nullptr, nullptr, xr);

    // ---- attention: logits, segment softmax, scatter-sum ----
    int nthr = Et * 8;
    edge_logits_kernel<<<(nthr + 255) / 256, 256, 0, stream>>>(ei, xl, xr, att, exbuf, maxbuf, E, Et);
    edge_exp_kernel<<<(nthr + 255) / 256, 256, 0, stream>>>(ei, exbuf, maxbuf, denom, E, Et);
    edge_scatter_kernel<<<(nthr + 255) / 256, 256, 0, stream>>>(ei, xl, exbuf, denom, Hacc, E, Et);

    // ---- relu + conv_bias + BN statistics ----
    relu_bn_stats_kernel<<<512, 128, 0, stream>>>(Hacc, conv_bias, sumb, sumsq, N);
    bn_final_kernel<<<1, 128, 0, stream>>>(sumb, sumsq, bn_gamma, bn_beta, scaleb, shiftb, N);

    // ---- z = BN(Hfeat)@lin_W + lin_b (BN fused into A-load) ----
    gemm128_wmma<<<rowTiles, 256, 0, stream>>>(Hacc, lin_W, lin_b, scaleb, shiftb, z);

    // ---- value2 = z@dec2_W + dec2_b ----
    gemm128_wmma<<<rowTiles, 256, 0, stream>>>(z, dec2_W, dec2_b, nullptr, nullptr, val2);

    // ---- value1 = sigmoid(-(relu(a)*dist + b)) over original edges ----
    int ndec = E * 8;
    edge_dec_kernel<<<(ndec + 255) / 256, 256, 0, stream>>>(ei, z, a_edge, b_edge, val1, E);
}